// MaskLayer_3032246911045
// MI455X (gfx1250) — compile-verified
//
#include <hip/hip_runtime.h>
#include <stdint.h>

// ---------------------------------------------------------------------------
// MaskLayer for MI455X (gfx1250).
// Memory-bound: ~96 MiB of traffic -> ~4.2us floor @ 23.3 TB/s. No matmul, so
// the CDNA5 path exercised here is ASYNC global->LDS 128b staging (ASYNCcnt)
// plus non-temporal streaming stores. Mask generation replicates JAX
// threefry2x32 with the reference's fixed key(42).
// ---------------------------------------------------------------------------

typedef float v4f __attribute__((ext_vector_type(4)));
typedef int   v4i __attribute__((ext_vector_type(4)));

#define AS1 __attribute__((address_space(1)))
#define AS3 __attribute__((address_space(3)))

#if __has_builtin(__builtin_amdgcn_global_load_async_to_lds_b128)
#define HAVE_ASYNC_LDS 1
#else
#define HAVE_ASYNC_LDS 0
#endif

static constexpr int B = 8, C = 128, H = 64, W = 1024;
static constexpr long long NELEM = (long long)B * C * H * W; // 8388608

// ----------------------------- threefry2x32 --------------------------------
__device__ __forceinline__ uint32_t rotl32(uint32_t v, uint32_t r) {
  return (v << r) | (v >> (32u - r));
}

__device__ __forceinline__ void tf_round(uint32_t &x0, uint32_t &x1, uint32_t r) {
  x0 += x1; x1 = rotl32(x1, r); x1 ^= x0;
}

__device__ void threefry2x32(uint32_t k0, uint32_t k1, uint32_t x0, uint32_t x1,
                             uint32_t &o0, uint32_t &o1) {
  const uint32_t ks0 = k0, ks1 = k1, ks2 = k0 ^ k1 ^ 0x1BD11BDAu;
  x0 += ks0; x1 += ks1;
  tf_round(x0, x1, 13); tf_round(x0, x1, 15); tf_round(x0, x1, 26); tf_round(x0, x1, 6);
  x0 += ks1; x1 += ks2 + 1u;
  tf_round(x0, x1, 17); tf_round(x0, x1, 29); tf_round(x0, x1, 16); tf_round(x0, x1, 24);
  x0 += ks2; x1 += ks0 + 2u;
  tf_round(x0, x1, 13); tf_round(x0, x1, 15); tf_round(x0, x1, 26); tf_round(x0, x1, 6);
  x0 += ks0; x1 += ks1 + 3u;
  tf_round(x0, x1, 17); tf_round(x0, x1, 29); tf_round(x0, x1, 16); tf_round(x0, x1, 24);
  x0 += ks1; x1 += ks2 + 4u;
  tf_round(x0, x1, 13); tf_round(x0, x1, 15); tf_round(x0, x1, 26); tf_round(x0, x1, 6);
  x0 += ks2; x1 += ks0 + 5u;
  o0 = x0; o1 = x1;
}

__device__ __forceinline__ float uniform01(uint32_t bits) {
  return __uint_as_float((bits >> 9) | 0x3f800000u) - 1.0f;
}

// ------------------------- mask generation kernel --------------------------
// One block, 256 threads. Builds mask_t[8][1024] (span 6) and mask_c[8][64]
// (span 1) exactly as jax.random with key(42) would, writes float masks to
// d_out tail and packed bitmask words to d_ws.
__global__ __launch_bounds__(256) void mask_gen_kernel(
    float* __restrict__ mask_t_out, float* __restrict__ mask_c_out,
    uint32_t* __restrict__ wst, uint32_t* __restrict__ wsc) {
  __shared__ uint32_t sw_t[256]; // [8 rows][32 words] seed bits for W=1024
  __shared__ uint32_t sw_c[16];  // [8 rows][2 words]  seed bits for H=64
  const int tid = threadIdx.x;
  sw_t[tid] = 0u;
  if (tid < 16) sw_c[tid] = 0u;
  __syncthreads();

  // key(42) -> data (0, 42); split -> kt, kc
  uint32_t a0, b0, a1, b1;
  threefry2x32(0u, 42u, 0u, 2u, a0, b0);
  threefry2x32(0u, 42u, 1u, 3u, a1, b1);
  const uint32_t kt0 = a0, kt1 = a1, kc0 = b0, kc1 = b1;
  // split(kt) -> k1t (uniform), k2t (randint)
  uint32_t c0, d0, c1, d1;
  threefry2x32(kt0, kt1, 0u, 2u, c0, d0);
  threefry2x32(kt0, kt1, 1u, 3u, c1, d1);
  const uint32_t k1t0 = c0, k1t1 = c1, k2t0 = d0, k2t1 = d1;
  // split(kc) -> k1c, k2c
  uint32_t e0, f0, e1, f1;
  threefry2x32(kc0, kc1, 0u, 2u, e0, f0);
  threefry2x32(kc0, kc1, 1u, 3u, e1, f1);
  const uint32_t k1c0 = e0, k1c1 = e1, k2c0 = f0, k2c1 = f1;

  // time-mask seeds: 8192 uniforms -> 4096 threefry lanes (i, i+4096)
  for (int l = tid; l < 4096; l += 256) {
    uint32_t r0, r1;
    threefry2x32(k1t0, k1t1, (uint32_t)l, (uint32_t)(l + 4096), r0, r1);
    if (uniform01(r0) < 0.1f) atomicOr(&sw_t[l >> 5], 1u << (l & 31));
    const int p = l + 4096;
    if (uniform01(r1) < 0.1f) atomicOr(&sw_t[p >> 5], 1u << (p & 31));
  }
  // channel-mask seeds: 512 uniforms -> 256 lanes (i, i+256)
  {
    uint32_t r0, r1;
    threefry2x32(k1c0, k1c1, (uint32_t)tid, (uint32_t)(tid + 256), r0, r1);
    if (uniform01(r0) < 0.1f) atomicOr(&sw_c[tid >> 5], 1u << (tid & 31));
    const int p = tid + 256;
    if (uniform01(r1) < 0.1f) atomicOr(&sw_c[p >> 5], 1u << (p & 31));
  }
  __syncthreads();

  // forced seed for seedless rows. jax randint draws bits shape (2,B);
  // span is a power of two so (2^32 % span)==0 and offset == lower & (L-1).
  if (tid < 8) {
    uint32_t any = 0;
    for (int w = 0; w < 32; ++w) any |= sw_t[tid * 32 + w];
    if (!any) {
      uint32_t r0, r1;
      threefry2x32(k2t0, k2t1, (uint32_t)tid, (uint32_t)(tid + 8), r0, r1);
      const uint32_t pos = r1 & 1023u;
      atomicOr(&sw_t[tid * 32 + (pos >> 5)], 1u << (pos & 31));
    }
  } else if (tid < 16) {
    const int b = tid - 8;
    const uint32_t any = sw_c[b * 2] | sw_c[b * 2 + 1];
    if (!any) {
      uint32_t r0, r1;
      threefry2x32(k2c0, k2c1, (uint32_t)b, (uint32_t)(b + 8), r0, r1);
      const uint32_t pos = r1 & 63u;
      atomicOr(&sw_c[b * 2 + (pos >> 5)], 1u << (pos & 31));
    }
  }
  __syncthreads();

  // span-6 extension, bit-parallel: mask[j] = OR_{s=0..5} seed[j-s]
  {
    const uint32_t cur = sw_t[tid];
    const uint32_t prev = (tid & 31) ? sw_t[tid - 1] : 0u; // no carry across rows
    uint32_t m = cur;
#pragma unroll
    for (int s = 1; s < 6; ++s) m |= (cur << s) | (prev >> (32 - s));
    wst[tid] = m;
    float* o = mask_t_out + tid * 32;
#pragma unroll
    for (int j = 0; j < 32; ++j) o[j] = ((m >> j) & 1u) ? 1.0f : 0.0f;
  }
  if (tid < 16) {
    const uint32_t m = sw_c[tid]; // span==1
    wsc[tid] = m;
    float* o = mask_c_out + tid * 32;
#pragma unroll
    for (int j = 0; j < 32; ++j) o[j] = ((m >> j) & 1u) ? 1.0f : 0.0f;
  }
}

// --------------------------- mask apply kernel -----------------------------
// One block per (b,c,h) row of 1024 floats. Each lane async-stages its own
// 16B into LDS (ASYNCcnt path), waits per-wave, selects, and streams out two
// non-temporal b128 stores. out = cbit ? c_rep : (tbit ? t_rep : x).
__global__ __launch_bounds__(256) void apply_mask_kernel(
    const float* __restrict__ x, const float* __restrict__ trep,
    const float* __restrict__ crep, const uint32_t* __restrict__ wst,
    const uint32_t* __restrict__ wsc, float* __restrict__ out,
    float* __restrict__ xcopy) {
  __shared__ v4f stage[256];
  const int t = threadIdx.x;
  const int row = blockIdx.x;      // 65536 rows = B*C*H
  const int b = row >> 13;         // C*H = 8192
  const int c = (row >> 6) & 127;
  const int h = row & 63;
  const long long base = (long long)row * 1024 + (long long)t * 4;

  v4f v;
#if HAVE_ASYNC_LDS
  __builtin_amdgcn_global_load_async_to_lds_b128(
      (AS1 v4i*)(x + base), (AS3 v4i*)&stage[t], 0, 0);
#if __has_builtin(__builtin_amdgcn_s_wait_asynccnt)
  __builtin_amdgcn_s_wait_asynccnt(0);
#else
  asm volatile("s_wait_asynccnt 0x0" ::: "memory");
#endif
  asm volatile("" ::: "memory");
  v = stage[t]; // each lane reads exactly the bytes it staged -> no barrier
#else
  v = *(const v4f*)(x + base);
#endif

  const float tr = trep[c];
  const float cr = crep[c];
  const uint32_t cbit = (wsc[b * 2 + (h >> 5)] >> (h & 31)) & 1u;
  const uint32_t tb = wst[b * 32 + (t >> 3)] >> ((t * 4) & 31);

  v4f o;
  o.x = cbit ? cr : ((tb & 1u) ? tr : v.x);
  o.y = cbit ? cr : ((tb & 2u) ? tr : v.y);
  o.z = cbit ? cr : ((tb & 4u) ? tr : v.z);
  o.w = cbit ? cr : ((tb & 8u) ? tr : v.w);

  __builtin_nontemporal_store(o, (v4f*)(out + base));
  __builtin_nontemporal_store(v, (v4f*)(xcopy + base));
}

// ------------------------------- launcher ----------------------------------
extern "C" void kernel_launch(void* const* d_in, const int* in_sizes, int n_in,
                              void* d_out, int out_size, void* d_ws,
                              size_t ws_size, hipStream_t stream) {
  const float* x = (const float*)d_in[0];
  const float* trep = (const float*)d_in[1];
  const float* crep = (const float*)d_in[2];

  float* out = (float*)d_out;
  float* xcopy = out + NELEM;
  float* mask_t_out = xcopy + NELEM;          // [8,1024]
  float* mask_c_out = mask_t_out + B * W;     // [8,64]

  uint32_t* wst = (uint32_t*)d_ws; // 256 words: mask_t bits [8][32]
  uint32_t* wsc = wst + 256;       // 16 words:  mask_c bits [8][2]

  mask_gen_kernel<<<1, 256, 0, stream>>>(mask_t_out, mask_c_out, wst, wsc);
  apply_mask_kernel<<<B * C * H, 256, 0, stream>>>(x, trep, crep, wst, wsc, out,
                                                   xcopy);
}